// SwinDAttention_29652454211859
// MI455X (gfx1250) — compile-verified
//
#include <hip/hip_runtime.h>
#include <hip/hip_bf16.h>

typedef float v2f __attribute__((ext_vector_type(2)));
typedef float v8f __attribute__((ext_vector_type(8)));

#define B_WIN   64
#define DIM     256
#define NPOS    256           // 16x16 positions per window
#define NH      8
#define HC      32
#define NG      4
#define GC      64
#define WIN_ELEMS (DIM * NPOS)  // 65536 floats per window

__device__ __forceinline__ v8f wmma_f32(v2f a, v2f b, v8f c) {
    // V_WMMA_F32_16X16X4_F32: D = A(16x4) * B(4x16) + C(16x16), fp32 throughout
    return __builtin_amdgcn_wmma_f32_16x16x4_f32(
        /*neg_a=*/false, a, /*neg_b=*/false, b,
        /*c_mod=*/(short)0, c, /*reuse_a=*/false, /*reuse_b=*/false);
}

// ---------------------------------------------------------------------------
// Generic per-window conv1x1 GEMM:  Y[o,n] = sum_c W[o,c] * X[b][n][c] + bias[o]
// X layout: [b][n][c] (c contiguous) -> B operand is one contiguous float2/lane.
// W layout: [o][c] row-major        -> A operand is one contiguous float2/lane.
// storeTrans=1 -> Y[b][n][o] ; storeTrans=0 -> Y[b][o][n]
// grid = 64 windows * 16 M-tiles, block = 128 threads (4 waves),
// wave = 16x64 output strip (4 accumulator tiles; A operand amortized 4x).
// ---------------------------------------------------------------------------
__global__ __launch_bounds__(128) void proj_gemm(
    const float* __restrict__ W, const float* __restrict__ bias,
    const float* __restrict__ X, float* __restrict__ Y, int storeTrans)
{
    const int blk   = blockIdx.x;
    const int b     = blk >> 4;
    const int Mbase = (blk & 15) << 4;
    const int lane  = threadIdx.x & 31;
    const int wave  = threadIdx.x >> 5;      // 0..3
    const int mrow  = lane & 15;
    const int khalf = lane >> 4;

    const float* Wr = W + (Mbase + mrow) * DIM + 2 * khalf;
    const int Nb    = wave * 64;             // this wave covers N in [Nb, Nb+64)
    const float* X0 = X + b * WIN_ELEMS + (Nb + mrow) * DIM + 2 * khalf;

    v8f c0, c1, c2, c3;
#pragma unroll
    for (int r = 0; r < 8; ++r) {
        float bv = bias[Mbase + r + 8 * khalf];
        c0[r] = bv; c1[r] = bv; c2[r] = bv; c3[r] = bv;
    }

#pragma unroll 4
    for (int kk = 0; kk < DIM; kk += 4) {
        v2f a  = *(const v2f*)(Wr + kk);
        v2f b0 = *(const v2f*)(X0 + kk);
        v2f b1 = *(const v2f*)(X0 + kk + 16 * DIM);
        v2f b2 = *(const v2f*)(X0 + kk + 32 * DIM);
        v2f b3 = *(const v2f*)(X0 + kk + 48 * DIM);
        c0 = wmma_f32(a, b0, c0);
        c1 = wmma_f32(a, b1, c1);
        c2 = wmma_f32(a, b2, c2);
        c3 = wmma_f32(a, b3, c3);
    }

    float* Yb = Y + b * WIN_ELEMS;
#pragma unroll
    for (int r = 0; r < 8; ++r) {
        const int o = Mbase + r + 8 * khalf;
        if (storeTrans) {
            Yb[(Nb + mrow) * DIM + o]      = c0[r];
            Yb[(Nb + 16 + mrow) * DIM + o] = c1[r];
            Yb[(Nb + 32 + mrow) * DIM + o] = c2[r];
            Yb[(Nb + 48 + mrow) * DIM + o] = c3[r];
        } else {
            Yb[o * DIM + Nb + mrow]        = c0[r];
            Yb[o * DIM + Nb + 16 + mrow]   = c1[r];
            Yb[o * DIM + Nb + 32 + mrow]   = c2[r];
            Yb[o * DIM + Nb + 48 + mrow]   = c3[r];
        }
    }
}

// ---------------------------------------------------------------------------
// Offset head: depthwise 3x3 -> channel LayerNorm -> GELU -> 1x1(->2) -> tanh
// One wave per (group-window, pixel). q layout: [b][hw][o].
// Writes pos (ws + d_out) and ref (d_out).
// ---------------------------------------------------------------------------
__global__ __launch_bounds__(256) void offset_kernel(
    const float* __restrict__ q, const float* __restrict__ dww,
    const float* __restrict__ dwb, const float* __restrict__ ln_g,
    const float* __restrict__ ln_b, const float* __restrict__ offw,
    float* __restrict__ posws, float* __restrict__ out_pos,
    float* __restrict__ out_ref)
{
    const int lane = threadIdx.x & 31;
    const int wave = threadIdx.x >> 5;
    const int p    = blockIdx.x * 8 + wave;   // 0..65535
    const int bg   = p >> 8;
    const int pix  = p & 255;
    const int b    = bg >> 2;
    const int g    = bg & 3;
    const int y    = pix >> 4;
    const int x    = pix & 15;

    const float* qb = q + b * WIN_ELEMS + g * GC;

    float v[2];
#pragma unroll
    for (int j = 0; j < 2; ++j) {
        const int c = lane + 32 * j;
        float acc = dwb[c];
#pragma unroll
        for (int ky = 0; ky < 3; ++ky) {
            int yy = y + ky - 1;
            if (yy < 0 || yy > 15) continue;
#pragma unroll
            for (int kx = 0; kx < 3; ++kx) {
                int xx = x + kx - 1;
                if (xx < 0 || xx > 15) continue;
                acc += dww[c * 9 + ky * 3 + kx] * qb[(yy * 16 + xx) * DIM + c];
            }
        }
        v[j] = acc;
    }

    float s  = v[0] + v[1];
    float ss = v[0] * v[0] + v[1] * v[1];
#pragma unroll
    for (int m = 16; m >= 1; m >>= 1) {
        s  += __shfl_xor(s,  m, 32);
        ss += __shfl_xor(ss, m, 32);
    }
    const float mean = s * (1.0f / 64.0f);
    const float var  = ss * (1.0f / 64.0f) - mean * mean;
    const float inv  = rsqrtf(var + 1e-5f);

    float py = 0.f, px = 0.f;
#pragma unroll
    for (int j = 0; j < 2; ++j) {
        const int c = lane + 32 * j;
        float hn = (v[j] - mean) * inv * ln_g[c] + ln_b[c];
        float ge = 0.5f * hn * (1.0f + erff(hn * 0.70710678118654752f));
        py += ge * offw[c];
        px += ge * offw[GC + c];
    }
#pragma unroll
    for (int m = 16; m >= 1; m >>= 1) {
        py += __shfl_xor(py, m, 32);
        px += __shfl_xor(px, m, 32);
    }

    if (lane == 0) {
        // tanh(off) * (1/16) * 2.0
        float offy = tanhf(py) * 0.125f;
        float offx = tanhf(px) * 0.125f;
        float ry = (y + 0.5f) * 0.125f - 1.0f;
        float rx = (x + 0.5f) * 0.125f - 1.0f;
        float pyv = offy + ry;
        float pxv = offx + rx;
        int o = bg * 512 + pix * 2;
        posws[o]     = pyv;  posws[o + 1]   = pxv;
        out_pos[o]   = pyv;  out_pos[o + 1] = pxv;
        out_ref[o]   = ry;   out_ref[o + 1] = rx;
    }
}

// ---------------------------------------------------------------------------
// Bilinear sampling of x (per-group channels) at deformed positions.
// One 64-thread block per (bg, sample n); thread = group channel (coalesced).
// xs layout: [b][n][c].
// ---------------------------------------------------------------------------
__global__ __launch_bounds__(64) void sample_kernel(
    const float* __restrict__ x, const float* __restrict__ posws,
    float* __restrict__ xs)
{
    const int bgn = blockIdx.x;
    const int bg  = bgn >> 8;
    const int n   = bgn & 255;
    const int b   = bg >> 2;
    const int g   = bg & 3;
    const int c   = threadIdx.x;

    const float py = posws[bg * 512 + n * 2];
    const float px = posws[bg * 512 + n * 2 + 1];
    const float gx = (px + 1.0f) * 7.5f;   // (x+1)*0.5*(W-1)
    const float gy = (py + 1.0f) * 7.5f;
    const float x0f = floorf(gx), y0f = floorf(gy);
    const float wx1 = gx - x0f, wx0 = 1.0f - wx1;
    const float wy1 = gy - y0f, wy0 = 1.0f - wy1;

    const float* xb = x + b * WIN_ELEMS + g * GC + c;
    float acc = 0.0f;
#pragma unroll
    for (int dy = 0; dy < 2; ++dy) {
#pragma unroll
        for (int dx = 0; dx < 2; ++dx) {
            float yf = y0f + dy, xf = x0f + dx;
            if (yf >= 0.0f && yf <= 15.0f && xf >= 0.0f && xf <= 15.0f) {
                int yi = (int)yf, xi = (int)xf;
                float w = (dy ? wy1 : wy0) * (dx ? wx1 : wx0);
                acc += w * xb[(yi * 16 + xi) * DIM];
            }
        }
    }
    xs[b * WIN_ELEMS + n * DIM + g * GC + c] = acc;
}

// ---------------------------------------------------------------------------
// Fused attention per (window, head): attn = scale*QtK + rpe-bias; softmax;
// out = attn*V. 512 threads (16 waves), each wave owns 16 query rows.
// Full 256x256 attention strip kept in LDS (~306 KB of the 320 KB WGP LDS).
// V strip staged LDS-direct via GLOBAL_LOAD_ASYNC_TO_LDS_B128 (ASYNCcnt).
// ---------------------------------------------------------------------------
#define ATTN_STRIDE 260   // 260 % 64 == 4 -> conflict-free WMMA-A b64 reads
#define VS_STRIDE   40    // 2*40 % 64 == 16 -> half-waves hit disjoint banks
#define SMEM_FLOATS (256 * ATTN_STRIDE + 256 * VS_STRIDE + 968 + 512)

__global__ __launch_bounds__(512) void attn_kernel(
    const float* __restrict__ q, const float* __restrict__ k,
    const float* __restrict__ v, const float* __restrict__ posws,
    const float* __restrict__ rpe, float* __restrict__ outb)
{
    extern __shared__ float smem[];
    float* attnS = smem;                          // [256][ATTN_STRIDE]
    float* vS    = smem + 256 * ATTN_STRIDE;      // [256][VS_STRIDE]
    float* rpeS  = vS + 256 * VS_STRIDE;          // 31*31 (+pad)
    float* posS  = rpeS + 968;                    // 256*2

    const int bh   = blockIdx.x;
    const int b    = bh >> 3;
    const int h    = bh & 7;
    const int tid  = threadIdx.x;
    const int lane = tid & 31;
    const int wave = tid >> 5;

    // ---- async DMA: V strip (256 rows x 128B) straight into padded LDS ----
    {
        const uint32_t vS_base = (uint32_t)(256u * ATTN_STRIDE) * 4u; // byte off of vS
        const float* vg = v + b * WIN_ELEMS + h * HC;
#pragma unroll
        for (int it = 0; it < 4; ++it) {
            const int i  = tid + it * 512;         // 0..2047
            const int n  = i >> 3;                 // V row
            const int ch = i & 7;                  // 16B chunk within row
            const uint32_t lds_off = vS_base + (uint32_t)(n * VS_STRIDE + ch * 4) * 4u;
            const float* gp = vg + n * DIM + ch * 4;
            asm volatile("global_load_async_to_lds_b128 %0, %1, off"
                         :: "v"(lds_off), "v"(gp) : "memory");
        }
    }

    // ---- stage rpe table and pos with plain loads meanwhile ----
    for (int i = tid; i < 961; i += 512) rpeS[i] = rpe[h * 961 + i];
    {
        const int bg = b * 4 + (h >> 1);
        posS[tid] = posws[bg * 512 + tid];
    }
    asm volatile("s_wait_asynccnt 0x0" ::: "memory");
    __syncthreads();

    const int Mbase = wave * 16;
    const int mrow  = lane & 15;
    const int khalf = lane >> 4;
    const float qgy = (wave + 0.5f) * 0.125f - 1.0f;   // row block y-coord (m>>4 == wave)

    const float* qrow = q + b * WIN_ELEMS + (Mbase + mrow) * DIM + h * HC + 2 * khalf;

    // ---- phase 1: attn tile = scale * Q^T K + bias(rpe bilinear) ----
    for (int nt = 0; nt < 16; ++nt) {
        const int Nbase = nt * 16;
        const float* krow = k + b * WIN_ELEMS + (Nbase + mrow) * DIM + h * HC + 2 * khalf;
        v8f c;
#pragma unroll
        for (int r = 0; r < 8; ++r) c[r] = 0.0f;
#pragma unroll
        for (int kk = 0; kk < HC; kk += 4) {
            v2f a  = *(const v2f*)(qrow + kk);
            v2f bb = *(const v2f*)(krow + kk);
            c = wmma_f32(a, bb, c);
        }
        const int n  = Nbase + mrow;
        const float posy = posS[n * 2];
        const float posx = posS[n * 2 + 1];
#pragma unroll
        for (int r = 0; r < 8; ++r) {
            const int m = Mbase + r + 8 * khalf;
            float val = c[r] * 0.17677669529663689f;    // 32^-0.5
            // bias = bilinear sample of rpe at disp = (qg - pos)*0.5 (zero pad)
            float qgx = ((m & 15) + 0.5f) * 0.125f - 1.0f;
            float dyc = (qgy - posy) * 0.5f;
            float dxc = (qgx - posx) * 0.5f;
            float gx = (dxc + 1.0f) * 15.0f;            // (d+1)*0.5*30
            float gy = (dyc + 1.0f) * 15.0f;
            float x0f = floorf(gx), y0f = floorf(gy);
            float wx1 = gx - x0f, wx0 = 1.0f - wx1;
            float wy1 = gy - y0f, wy0 = 1.0f - wy1;
            float bias = 0.0f;
#pragma unroll
            for (int dy = 0; dy < 2; ++dy) {
#pragma unroll
                for (int dx = 0; dx < 2; ++dx) {
                    float yf = y0f + dy, xf = x0f + dx;
                    if (yf >= 0.0f && yf <= 30.0f && xf >= 0.0f && xf <= 30.0f) {
                        float w = (dy ? wy1 : wy0) * (dx ? wx1 : wx0);
                        bias += w * rpeS[(int)yf * 31 + (int)xf];
                    }
                }
            }
            attnS[m * ATTN_STRIDE + n] = val + bias;
        }
    }
    __syncthreads();

    // ---- phase 2: softmax over n (each wave owns its 16 rows) ----
    for (int r = 0; r < 16; ++r) {
        const int m = Mbase + r;
        float vals[8];
        float mx = -3.4e38f;
#pragma unroll
        for (int j = 0; j < 8; ++j) {
            vals[j] = attnS[m * ATTN_STRIDE + lane + 32 * j];
            mx = fmaxf(mx, vals[j]);
        }
#pragma unroll
        for (int mm = 16; mm >= 1; mm >>= 1) mx = fmaxf(mx, __shfl_xor(mx, mm, 32));
        float sum = 0.0f;
#pragma unroll
        for (int j = 0; j < 8; ++j) { vals[j] = __expf(vals[j] - mx); sum += vals[j]; }
#pragma unroll
        for (int mm = 16; mm >= 1; mm >>= 1) sum += __shfl_xor(sum, mm, 32);
        const float invs = 1.0f / sum;
#pragma unroll
        for (int j = 0; j < 8; ++j) attnS[m * ATTN_STRIDE + lane + 32 * j] = vals[j] * invs;
    }
    __syncthreads();

    // ---- phase 3: out = P * V  (A from LDS attention strip, B from LDS V) ----
    float* outw = outb + b * WIN_ELEMS + h * HC;
#pragma unroll
    for (int ct = 0; ct < 2; ++ct) {
        const int Cbase = ct * 16;
        v8f c;
#pragma unroll
        for (int r = 0; r < 8; ++r) c[r] = 0.0f;
#pragma unroll 4
        for (int kk = 0; kk < NPOS; kk += 4) {
            v2f a = *(const v2f*)(attnS + (Mbase + mrow) * ATTN_STRIDE + kk + 2 * khalf);
            const int kr = kk + 2 * khalf;
            v2f bb;
            bb.x = vS[kr * VS_STRIDE + Cbase + mrow];
            bb.y = vS[(kr + 1) * VS_STRIDE + Cbase + mrow];
            c = wmma_f32(a, bb, c);
        }
#pragma unroll
        for (int r = 0; r < 8; ++r) {
            const int m = Mbase + r + 8 * khalf;
            outw[m * DIM + Cbase + mrow] = c[r];
        }
    }
}

// ---------------------------------------------------------------------------
extern "C" void kernel_launch(void* const* d_in, const int* in_sizes, int n_in,
                              void* d_out, int out_size, void* d_ws, size_t ws_size,
                              hipStream_t stream) {
    (void)in_sizes; (void)n_in; (void)out_size; (void)ws_size;

    const float* x    = (const float*)d_in[0];
    const float* qw   = (const float*)d_in[1];
    const float* qb   = (const float*)d_in[2];
    const float* kw   = (const float*)d_in[3];
    const float* kb   = (const float*)d_in[4];
    const float* vw   = (const float*)d_in[5];
    const float* vb   = (const float*)d_in[6];
    const float* ow   = (const float*)d_in[7];
    const float* ob   = (const float*)d_in[8];
    const float* dww  = (const float*)d_in[9];
    const float* dwb  = (const float*)d_in[10];
    const float* ln_g = (const float*)d_in[11];
    const float* ln_b = (const float*)d_in[12];
    const float* offw = (const float*)d_in[13];
    const float* rpe  = (const float*)d_in[14];

    float* ws    = (float*)d_ws;
    float* qbuf  = ws;                      // [64][256][256]  q, [b][n][o]
    float* xsb   = ws + 4194304;            // [64][256][256]  xs, [b][n][c]
    float* kbuf  = ws + 8388608;            // [64][256][256]  k,  [b][n][o]
    float* vbuf  = ws + 12582912;           // [64][256][256]  v,  [b][n][o]
    float* outbf = xsb;                     // alias: xs dead after k/v GEMMs
    float* posws = ws + 16777216;           // [256][256][2]

    float* out     = (float*)d_out;
    float* y_out   = out;                   // [64][256][16][16]
    float* pos_out = out + 4194304;         // [256][16][16][2]
    float* ref_out = out + 4325376;         // [256][16][16][2]

    proj_gemm<<<1024, 128, 0, stream>>>(qw, qb, x, qbuf, 1);
    offset_kernel<<<8192, 256, 0, stream>>>(qbuf, dww, dwb, ln_g, ln_b, offw,
                                            posws, pos_out, ref_out);
    sample_kernel<<<65536, 64, 0, stream>>>(x, posws, xsb);
    proj_gemm<<<1024, 128, 0, stream>>>(kw, kb, xsb, kbuf, 1);
    proj_gemm<<<1024, 128, 0, stream>>>(vw, vb, xsb, vbuf, 1);
    attn_kernel<<<512, 512, SMEM_FLOATS * sizeof(float), stream>>>(
        qbuf, kbuf, vbuf, posws, rpe, outbf);
    proj_gemm<<<1024, 128, 0, stream>>>(ow, ob, outbf, y_out, 0);
}